// Discriminator_18056042512603
// MI455X (gfx1250) — compile-verified
//
#include <hip/hip_runtime.h>

typedef __attribute__((ext_vector_type(2))) float v2f;
typedef __attribute__((ext_vector_type(8))) float v8f;

#define D_FEAT 128

// ---------------- zero fill (grid-stride) ----------------
__global__ void zero_f32(float* __restrict__ p, long n) {
  long i = (long)blockIdx.x * blockDim.x + threadIdx.x;
  long stride = (long)gridDim.x * blockDim.x;
  for (; i < n; i += stride) p[i] = 0.0f;
}

// ---------------- degree accumulation over edges ----------------
__global__ void deg_kernel(const int* __restrict__ dst, float* __restrict__ deg, int E) {
  int e = blockIdx.x * blockDim.x + threadIdx.x;
  if (e < E) atomicAdd(&deg[dst[e]], 1.0f);
}

// deg -> dinv in place (self loop adds 1, so deg_total >= 1 always)
__global__ void dinv_kernel(float* __restrict__ deg, int N) {
  int i = blockIdx.x * blockDim.x + threadIdx.x;
  if (i < N) {
    float d = deg[i] + 1.0f;           // + self loop
    deg[i] = rsqrtf(fmaxf(d, 1.0f));
  }
}

// ---------------- layer-1 GEMM: h = x @ W1 via V_WMMA_F32_16X16X4_F32 -------
// One wave per 16x16 output tile. blockDim = (32, 8): 8 col-tiles cover 128.
// A (16x4 f32): lane L<16 holds M=L, K={k,k+1}; lane L>=16 holds M=L-16, K={k+2,k+3}
// B (4x16 f32): same pattern indexed by N instead of M.
// C/D (16x16 f32, 8 VGPRs): vgpr r -> M = r + (lane<16 ? 0 : 8), N = lane&15.
__global__ void __launch_bounds__(256) gemm1_wmma(const float* __restrict__ x,
                                                  const float* __restrict__ W1,
                                                  float* __restrict__ h, int N) {
  const int lane  = threadIdx.x;        // 0..31
  const int half  = lane >> 4;          // 0 or 1
  const int l15   = lane & 15;
  const int mBase = blockIdx.x * 16;
  const int nBase = threadIdx.y * 16;

  int mrow = mBase + l15;
  if (mrow > N - 1) mrow = N - 1;       // clamp keeps EXEC all-ones for WMMA

  const float* arow = x  + (long)mrow * D_FEAT + half * 2;
  const float* bcol = W1 + (long)(half * 2) * D_FEAT + nBase + l15;

  v8f acc = {};
#pragma unroll
  for (int k = 0; k < D_FEAT; k += 4) {
    v2f a, b;
    a.x = arow[k];
    a.y = arow[k + 1];
    b.x = bcol[(long)k * D_FEAT];
    b.y = bcol[(long)(k + 1) * D_FEAT];
    acc = __builtin_amdgcn_wmma_f32_16x16x4_f32(
        /*neg_a=*/false, a, /*neg_b=*/false, b,
        /*c_mod=*/(short)0, acc, /*reuse_a=*/false, /*reuse_b=*/false);
  }

#pragma unroll
  for (int r = 0; r < 8; ++r) {
    int row = mBase + r + half * 8;
    if (row < N) h[(long)row * D_FEAT + nBase + l15] = acc[r];
  }
}

// ---------------- layer-1 edge aggregation (L2-resident scatter-add) --------
// One wave per edge; each lane handles 4 of the 128 features (float4 gather,
// 4 scalar f32 atomic adds). h and agg both fit in the 192MB L2.
__global__ void __launch_bounds__(256) scatter1(const int* __restrict__ src,
                                                const int* __restrict__ dst,
                                                const float* __restrict__ dinv,
                                                const float* __restrict__ h,
                                                float* __restrict__ agg, int E) {
  long tid  = (long)blockIdx.x * blockDim.x + threadIdx.x;
  int  e    = (int)(tid >> 5);
  int  lane = (int)(tid & 31);
  if (e >= E) return;
  int s = src[e], d = dst[e];
  float nrm = dinv[s] * dinv[d];
  const float4 hv = *(const float4*)(h + (long)s * D_FEAT + lane * 4);
  float* ap = agg + (long)d * D_FEAT + lane * 4;
  atomicAdd(ap + 0, hv.x * nrm);
  atomicAdd(ap + 1, hv.y * nrm);
  atomicAdd(ap + 2, hv.z * nrm);
  atomicAdd(ap + 3, hv.w * nrm);
}

// ---------------- fused: self-loop + bias + relu + (. @ W2) + layer-2 self --
// One wave per node. z[i] = relu(agg[i] + h[i]*dinv^2 + b1) . W2
// out[i] initialized to b2 + z[i]*dinv[i]^2 (layer-2 self-loop message).
__global__ void __launch_bounds__(256) node_combine(const float* __restrict__ h,
                                                    const float* __restrict__ agg,
                                                    const float* __restrict__ dinv,
                                                    const float* __restrict__ b1,
                                                    const float* __restrict__ W2,
                                                    const float* __restrict__ b2,
                                                    float* __restrict__ z,
                                                    float* __restrict__ out, int N) {
  long tid  = (long)blockIdx.x * blockDim.x + threadIdx.x;
  int  node = (int)(tid >> 5);
  int  lane = (int)(tid & 31);
  if (node >= N) return;

  float di    = dinv[node];
  float selfn = di * di;
  long  base  = (long)node * D_FEAT + lane * 4;

  float4 hv = *(const float4*)(h   + base);
  float4 av = *(const float4*)(agg + base);
  float4 bv = *(const float4*)(b1 + lane * 4);
  float4 wv = *(const float4*)(W2 + lane * 4);

  float t0 = fmaxf(av.x + hv.x * selfn + bv.x, 0.0f);
  float t1 = fmaxf(av.y + hv.y * selfn + bv.y, 0.0f);
  float t2 = fmaxf(av.z + hv.z * selfn + bv.z, 0.0f);
  float t3 = fmaxf(av.w + hv.w * selfn + bv.w, 0.0f);
  float partial = t0 * wv.x + t1 * wv.y + t2 * wv.z + t3 * wv.w;

#pragma unroll
  for (int off = 16; off > 0; off >>= 1)
    partial += __shfl_down(partial, off, 32);

  if (lane == 0) {
    z[node]   = partial;
    out[node] = b2[0] + partial * selfn;  // bias + layer-2 self-loop term
  }
}

// ---------------- layer-2 edge aggregation (scalar) -------------------------
__global__ void scatter2(const int* __restrict__ src, const int* __restrict__ dst,
                         const float* __restrict__ dinv, const float* __restrict__ z,
                         float* __restrict__ out, int E) {
  int e = blockIdx.x * blockDim.x + threadIdx.x;
  if (e < E) {
    int s = src[e], d = dst[e];
    atomicAdd(&out[d], z[s] * dinv[s] * dinv[d]);
  }
}

extern "C" void kernel_launch(void* const* d_in, const int* in_sizes, int n_in,
                              void* d_out, int out_size, void* d_ws, size_t ws_size,
                              hipStream_t stream) {
  const float* x  = (const float*)d_in[0];
  const int*   ei = (const int*)d_in[1];   // integer input -> const int* (harness mapping)
  const float* W1 = (const float*)d_in[2];
  const float* b1 = (const float*)d_in[3];
  const float* W2 = (const float*)d_in[4];
  const float* b2 = (const float*)d_in[5];
  float* out = (float*)d_out;

  const int N = in_sizes[0] / D_FEAT;      // 50000
  const int E = in_sizes[1] / 2;           // 800000
  const int* src = ei;
  const int* dst = ei + E;

  // workspace carve-up (all offsets 16B aligned: N % 4 == 0)
  float* deg = (float*)d_ws;               // N floats, becomes dinv
  float* z   = deg + N;                    // N floats
  float* h   = z + N;                      // N*128 floats
  float* agg = h + (long)N * D_FEAT;       // N*128 floats

  // 1) zero accumulators (must happen every call: harness doesn't re-poison)
  zero_f32<<<2048, 256, 0, stream>>>(deg, (long)N);
  zero_f32<<<2048, 256, 0, stream>>>(agg, (long)N * D_FEAT);

  // 2) degrees -> dinv
  deg_kernel<<<(E + 255) / 256, 256, 0, stream>>>(dst, deg, E);
  dinv_kernel<<<(N + 255) / 256, 256, 0, stream>>>(deg, N);

  // 3) h = x @ W1  (fp32 WMMA, one wave per 16x16 tile, 8 col-tiles per block)
  dim3 gblock(32, 8);
  gemm1_wmma<<<(N + 15) / 16, gblock, 0, stream>>>(x, W1, h, N);

  // 4) layer-1 neighbor aggregation (one wave per edge)
  long s1_threads = (long)E * 32;
  scatter1<<<(unsigned)((s1_threads + 255) / 256), 256, 0, stream>>>(src, dst, deg, h, agg, E);

  // 5) fused self-loop + bias + relu + dot(W2) + out init (one wave per node)
  long nc_threads = (long)N * 32;
  node_combine<<<(unsigned)((nc_threads + 255) / 256), 256, 0, stream>>>(
      h, agg, deg, b1, W2, b2, z, out, N);

  // 6) layer-2 scalar aggregation
  scatter2<<<(E + 255) / 256, 256, 0, stream>>>(src, dst, deg, z, out, E);
}